// GlobalFeatureBlock_Diffusion_85916525789882
// MI455X (gfx1250) — compile-verified
//
#include <hip/hip_runtime.h>
#include <stdint.h>

// Problem constants (match reference)
#define HH 56
#define WW 56
#define NP (HH*WW)          // 3136 elements per plane
#define NC 256              // channels
#define NB 32               // batch
#define NT 256              // threads per block (8 wave32)
#define NE 13               // ceil(NP/NT) elements per thread
#define NTAIL (NP - (NE-1)*NT)   // 64 active threads in last chunk
#define DT_C 0.2f
#define BN_EPS 1e-5f

// One explicit-Euler PDE step. RB = read buffer (compile-time so the
// +/-12544B plane select folds into the ds imm offset), WB = write back.
template<int RB, bool WB>
__device__ __forceinline__ void pde_step(
    float (&sH)[2][NP],
    float (&hr)[NE], const float (&fr)[NE],
    const float (&uh)[NE], const float (&vh)[NE],
    const float (&dxr)[NE], const float (&dyr)[NE], const float (&mds)[NE],
    const int (&iyp)[NE], const int (&iym)[NE],
    const int (&ixp)[NE], const int (&ixm)[NE],
    int tid)
{
    __syncthreads();
    #pragma unroll
    for (int k = 0; k < NE; ++k) {
        if (k < NE-1 || tid < NTAIL) {
            float h   = hr[k];
            float hxp = sH[RB][iyp[k]];     // roll -1 along H (periodic)
            float hxm = sH[RB][iym[k]];     // roll +1 along H
            float hyp = sH[RB][ixp[k]];     // roll -1 along W
            float hym = sH[RB][ixm[k]];     // roll +1 along W
            float sx  = hxp + hxm;
            float sy  = hyp + hym;
            float dx_ = hxp - hxm;
            float dy_ = hyp - hym;
            // rhs = f + Dx*(hxp+hxm) + Dy*(hyp+hym) - 2h(Dx+Dy) + 0.5u*dx + 0.5v*dy
            float rhs = fr[k];
            rhs = fmaf(dxr[k], sx,  rhs);
            rhs = fmaf(dyr[k], sy,  rhs);
            rhs = fmaf(uh[k],  dx_, rhs);
            rhs = fmaf(vh[k],  dy_, rhs);
            rhs = fmaf(mds[k], h,   rhs);
            hr[k] = fmaf(DT_C, rhs, h);
            if (WB) sH[RB ^ 1][tid + k*NT] = hr[k];  // imm-offset ds_store
        }
    }
}

__global__ __launch_bounds__(NT)
void pde_fused_kernel(const float* __restrict__ x,
                      const float* __restrict__ cw,    // [4,C,1,3,3]
                      const float* __restrict__ cbv,   // [4,C]
                      const float* __restrict__ bns,   // [5,C]
                      const float* __restrict__ bnb,   // [5,C]
                      const float* __restrict__ bnm,   // [5,C]
                      const float* __restrict__ bnv,   // [5,C]
                      float* __restrict__ out)
{
    __shared__ float sH[2][NP];              // 25 KB double-buffered h plane

    const int plane = blockIdx.x;            // = b*NC + c (planes contiguous)
    const int c     = plane & (NC - 1);
    const int tid   = threadIdx.x;
    const float* __restrict__ xp = x   + (size_t)plane * NP;
    float*       __restrict__ op = out + (size_t)plane * NP;

    // ---- per-channel scalars (uniform -> SGPRs / s_load) ----
    float wgt[4][9], bias[4];
    #pragma unroll
    for (int n = 0; n < 4; ++n) {
        bias[n] = cbv[n*NC + c];
        #pragma unroll
        for (int j = 0; j < 9; ++j) wgt[n][j] = cw[(n*NC + c)*9 + j];
    }
    float ga[5], gb[5];
    #pragma unroll
    for (int n = 0; n < 5; ++n) {
        float inv = bns[n*NC + c] * rsqrtf(bnv[n*NC + c] + BN_EPS);
        ga[n] = inv;
        gb[n] = bnb[n*NC + c] - bnm[n*NC + c]*inv;
    }

    // ---- async copy: x plane -> LDS sH[0] (784 x 16B chunks via ASYNCcnt) ----
    {
        uint32_t lbase = (uint32_t)(uintptr_t)(&sH[0][0]);   // LDS byte offset
        #pragma unroll
        for (int k = 0; k < 4; ++k) {
            int chunk = tid + k*NT;                          // NP/4 = 784
            if (chunk < NP/4) {
                uint32_t laddr = lbase + (uint32_t)chunk*16u;
                const float* g = xp + chunk*4;
                asm volatile("global_load_async_to_lds_b128 %0, %1, off"
                             :: "v"(laddr), "v"(g) : "memory");
            }
        }
        asm volatile("s_wait_asynccnt 0" ::: "memory");
        __syncthreads();
    }

    // ---- one-time: depthwise 3x3 convs (zero pad) + BN, coefficients to regs,
    //      plus hoisted periodic neighbor indices ----
    float fr[NE], hr[NE], uh[NE], vh[NE], dxr[NE], dyr[NE], mds[NE];
    int   iyp[NE], iym[NE], ixp[NE], ixm[NE];
    #pragma unroll
    for (int k = 0; k < NE; ++k) {
        int e = tid + k*NT;
        if (k < NE-1 || tid < NTAIL) {
            int y  = e / WW;
            int xw = e - y*WW;
            // periodic (roll) neighbor word-indices, loop-invariant over steps
            iyp[k] = ((y  == HH-1) ? 0    : y+1)*WW + xw;
            iym[k] = ((y  == 0)    ? HH-1 : y-1)*WW + xw;
            ixp[k] = y*WW + ((xw == WW-1) ? 0    : xw+1);
            ixm[k] = y*WW + ((xw == 0)    ? WW-1 : xw-1);

            float a0 = bias[0], a1 = bias[1], a2 = bias[2], a3 = bias[3];
            #pragma unroll
            for (int dy = -1; dy <= 1; ++dy) {
                #pragma unroll
                for (int dx = -1; dx <= 1; ++dx) {
                    int yy = y + dy, xx = xw + dx;
                    float val = 0.f;                         // zero padding
                    if (yy >= 0 && yy < HH && xx >= 0 && xx < WW)
                        val = sH[0][yy*WW + xx];
                    int t = (dy+1)*3 + (dx+1);
                    a0 = fmaf(wgt[0][t], val, a0);
                    a1 = fmaf(wgt[1][t], val, a1);
                    a2 = fmaf(wgt[2][t], val, a2);
                    a3 = fmaf(wgt[3][t], val, a3);
                }
            }
            float t0 = fmaf(a0, ga[0], gb[0]);
            float t1 = fmaf(a1, ga[1], gb[1]);
            float t2 = fmaf(a2, ga[2], gb[2]);
            float t3 = fmaf(a3, ga[3], gb[3]);
            uh[k]  = 0.5f * (t0 > 0.f ? t0 : 0.f);           // 0.5*relu(u)
            vh[k]  = 0.5f * (t1 > 0.f ? t1 : 0.f);           // 0.5*relu(v)
            float dxv = 1.f / (1.f + __expf(-t2));           // sigmoid, CDX=1
            float dyv = 1.f / (1.f + __expf(-t3));           // sigmoid, CDY=1
            dxr[k] = dxv;
            dyr[k] = dyv;
            mds[k] = -2.f * (dxv + dyv);
            float fv = sH[0][e];
            fr[k] = fv;
            hr[k] = fv;                                      // h0 = x
        } else {
            iyp[k] = iym[k] = ixp[k] = ixm[k] = 0;
            fr[k] = hr[k] = uh[k] = vh[k] = dxr[k] = dyr[k] = mds[k] = 0.f;
        }
    }

    // ---- 10 explicit Euler steps, ping-pong with compile-time buffer ids ----
    #pragma unroll 1
    for (int s2 = 0; s2 < 4; ++s2) {     // steps 0..7
        pde_step<0, true >(sH, hr, fr, uh, vh, dxr, dyr, mds, iyp, iym, ixp, ixm, tid);
        pde_step<1, true >(sH, hr, fr, uh, vh, dxr, dyr, mds, iyp, iym, ixp, ixm, tid);
    }
    pde_step<0, true >(sH, hr, fr, uh, vh, dxr, dyr, mds, iyp, iym, ixp, ixm, tid); // step 8
    pde_step<1, false>(sH, hr, fr, uh, vh, dxr, dyr, mds, iyp, iym, ixp, ixm, tid); // step 9 (regs only)

    // ---- final BN + relu, coalesced b32 stores ----
    #pragma unroll
    for (int k = 0; k < NE; ++k) {
        int e = tid + k*NT;
        if (k < NE-1 || tid < NTAIL) {
            float t = fmaf(hr[k], ga[4], gb[4]);
            op[e] = t > 0.f ? t : 0.f;
        }
    }
}

extern "C" void kernel_launch(void* const* d_in, const int* in_sizes, int n_in,
                              void* d_out, int out_size, void* d_ws, size_t ws_size,
                              hipStream_t stream) {
    const float* x   = (const float*)d_in[0];
    const float* cw  = (const float*)d_in[1];
    const float* cbv = (const float*)d_in[2];
    const float* bns = (const float*)d_in[3];
    const float* bnb = (const float*)d_in[4];
    const float* bnm = (const float*)d_in[5];
    const float* bnv = (const float*)d_in[6];
    float* out = (float*)d_out;

    pde_fused_kernel<<<dim3(NB*NC), dim3(NT), 0, stream>>>(
        x, cw, cbv, bns, bnb, bnm, bnv, out);

    (void)in_sizes; (void)n_in; (void)out_size; (void)d_ws; (void)ws_size;
}